// TripletMarginLoss_9088150798430
// MI455X (gfx1250) — compile-verified
//
#include <hip/hip_runtime.h>
#include <math.h>

#define MARGIN 0.3f
#define EPSF   1e-16f
#define MAXN   512

typedef __attribute__((ext_vector_type(2))) float v2f;
typedef __attribute__((ext_vector_type(8))) float v8f;

// ---------------------------------------------------------------------------
// Stage 1: squared norms per row  (N threads, D/4 float4 loads each)
// ---------------------------------------------------------------------------
__global__ __launch_bounds__(256) void sqnorm_kernel(const float* __restrict__ X,
                                                     float* __restrict__ sqn,
                                                     int N, int D) {
    int row = blockIdx.x * blockDim.x + threadIdx.x;
    if (row >= N) return;
    const float4* p = (const float4*)(X + (long)row * D);
    float acc = 0.f;
    for (int k = 0; k < (D >> 2); ++k) {
        float4 v = p[k];
        acc += v.x * v.x + v.y * v.y + v.z * v.z + v.w * v.w;
    }
    sqn[row] = acc;
}

// ---------------------------------------------------------------------------
// Stage 2: 16x16 tile of G = X X^T via V_WMMA_F32_16X16X4_F32, fused with the
// distance epilogue  d = sqrt(max(sqn_i + sqn_j - 2 G, 0)).
// One wave32 per tile; EXEC all-ones (WMMA requirement).
//
// fp32 A 16x4 layout (ISA 7.12.2): lane L -> M = L%16; VGPR v holds
// K = 2*(L>>4) + v.  B (4x16) mirrored with N in place of M.
// C/D 16x16 f32: VGPR r, lane L -> M = r + 8*(L>>4), N = L%16.
// ---------------------------------------------------------------------------
__global__ __launch_bounds__(32) void gram_dist_kernel(const float* __restrict__ X,
                                                       const float* __restrict__ sqn,
                                                       float* __restrict__ Dm,
                                                       int N, int D) {
    const int lane = threadIdx.x;            // 0..31
    const int i0 = blockIdx.y << 4;
    const int j0 = blockIdx.x << 4;
    const int m  = lane & 15;
    const int kh = (lane >> 4) << 1;         // 0 or 2: which K pair this half-wave owns

    const float* __restrict__ arow = X + (long)(i0 + m) * D + kh;
    const float* __restrict__ brow = X + (long)(j0 + m) * D + kh;

    v8f c = {};
    #pragma unroll 8
    for (int k = 0; k < D; k += 4) {
        v2f a = *(const v2f*)(arow + k);
        v2f b = *(const v2f*)(brow + k);
        // 8 args: (neg_a, A, neg_b, B, c_mod, C, reuse_a, reuse_b)
        c = __builtin_amdgcn_wmma_f32_16x16x4_f32(false, a, false, b,
                                                  (short)0, c, false, false);
    }

    const int   col  = j0 + m;
    const float sj   = sqn[col];
    const int   roff = (lane >> 4) << 3;     // 0 or 8
    #pragma unroll
    for (int r = 0; r < 8; ++r) {
        int   row = i0 + roff + r;
        float sq  = sqn[row] + sj - 2.0f * c[r];
        Dm[(long)row * N + col] = (sq > 0.f) ? sqrtf(sq) : 0.f;
    }
}

// ---------------------------------------------------------------------------
// Stage 3: one block per anchor i.  Stage distance-row + labels in LDS.
// Mask simplification: lab[j]==li && j!=i  (positive),  lab[k]!=li (negative)
// already implies k!=i and k!=j.
// ---------------------------------------------------------------------------
__global__ __launch_bounds__(256) void triplet_partial_kernel(const float* __restrict__ Dm,
                                                              const int* __restrict__ labels,
                                                              float* __restrict__ psum,
                                                              float* __restrict__ pcnt,
                                                              int N) {
    __shared__ float drow[MAXN];
    __shared__ int   lab[MAXN];
    __shared__ float rs[256];
    __shared__ float rc[256];

    const int i   = blockIdx.x;
    const int tid = threadIdx.x;

    for (int t = tid; t < N; t += 256) {
        drow[t] = Dm[(long)i * N + t];
        lab[t]  = labels[t];
    }
    __syncthreads();

    const int li = lab[i];
    float sum = 0.f, cnt = 0.f;

    for (int j = tid; j < N; j += 256) {
        if (j == i || lab[j] != li) continue;      // anchor-positive mask
        const float base = drow[j] + MARGIN;
        for (int k = 0; k < N; ++k) {
            if (lab[k] == li) continue;            // anchor-negative mask
            float v = base - drow[k];
            v = fmaxf(v, 0.f);
            sum += v;
            cnt += (v > EPSF) ? 1.f : 0.f;
        }
    }

    rs[tid] = sum;
    rc[tid] = cnt;
    __syncthreads();
    for (int s = 128; s > 0; s >>= 1) {
        if (tid < s) { rs[tid] += rs[tid + s]; rc[tid] += rc[tid + s]; }
        __syncthreads();
    }
    if (tid == 0) { psum[i] = rs[0]; pcnt[i] = rc[0]; }
}

// ---------------------------------------------------------------------------
// Stage 4: deterministic final reduction and divide.
// ---------------------------------------------------------------------------
__global__ __launch_bounds__(256) void finalize_kernel(const float* __restrict__ psum,
                                                       const float* __restrict__ pcnt,
                                                       float* __restrict__ out,
                                                       int N) {
    __shared__ float rs[256];
    __shared__ float rc[256];
    const int tid = threadIdx.x;
    float s = 0.f, c = 0.f;
    for (int t = tid; t < N; t += 256) { s += psum[t]; c += pcnt[t]; }
    rs[tid] = s;
    rc[tid] = c;
    __syncthreads();
    for (int k = 128; k > 0; k >>= 1) {
        if (tid < k) { rs[tid] += rs[tid + k]; rc[tid] += rc[tid + k]; }
        __syncthreads();
    }
    if (tid == 0) out[0] = rs[0] / (rc[0] + EPSF);
}

// ---------------------------------------------------------------------------
extern "C" void kernel_launch(void* const* d_in, const int* in_sizes, int n_in,
                              void* d_out, int out_size, void* d_ws, size_t ws_size,
                              hipStream_t stream) {
    const float* X      = (const float*)d_in[0];  // [N, D] fp32
    const int*   labels = (const int*)d_in[1];    // [N] int32
    float*       out    = (float*)d_out;          // scalar

    const int N = in_sizes[1];                    // 512
    const int D = in_sizes[0] / N;                // 128

    // Workspace layout (fp32): sqn[N] | Dm[N*N] | psum[N] | pcnt[N]  (~1.05 MB)
    float* sqn  = (float*)d_ws;
    float* Dm   = sqn + N;
    float* psum = Dm + (size_t)N * N;
    float* pcnt = psum + N;

    sqnorm_kernel<<<(N + 255) / 256, 256, 0, stream>>>(X, sqn, N, D);

    dim3 grid(N / 16, N / 16);
    gram_dist_kernel<<<grid, 32, 0, stream>>>(X, sqn, Dm, N, D);

    triplet_partial_kernel<<<N, 256, 0, stream>>>(Dm, labels, psum, pcnt, N);

    finalize_kernel<<<1, 256, 0, stream>>>(psum, pcnt, out, N);
}